// FARNN_37039797961222
// MI455X (gfx1250) — compile-verified
//
#include <hip/hip_runtime.h>

typedef float v2f __attribute__((ext_vector_type(2)));
typedef float v4f __attribute__((ext_vector_type(4)));
typedef float v8f __attribute__((ext_vector_type(8)));

#define LSEQ 512
#define KDIM 256
#define RDIM 32
#define H1D  100
#define H2D  100
#define NLAB 10

// ws layout (floats): [0..1023]  Bmat[k][n] = (D1@D2)[n][k]  (i.e. M^T, ready as WMMA B)
//                     [1024..1055] beta0[r] = alpha0 @ D1^T
__global__ __launch_bounds__(1024) void farnn_setup(
    const float* __restrict__ alpha0, const float* __restrict__ D1,
    const float* __restrict__ D2, float* __restrict__ ws) {
  int tid = threadIdx.x;                  // 1024 threads, one per M entry
  int k = tid >> 5, n = tid & 31;
  float s = 0.f;
  for (int c = 0; c < KDIM; ++c) s = fmaf(D1[n * KDIM + c], D2[c * RDIM + k], s);
  ws[k * RDIM + n] = s;
  if (tid < RDIM) {
    float b = 0.f;
    for (int c = 0; c < KDIM; ++c) b = fmaf(alpha0[c], D1[tid * KDIM + c], b);
    ws[RDIM * RDIM + tid] = b;
  }
}

static __device__ __forceinline__ v8f wmma_f32_4(v2f a, v2f b, v8f c) {
  return __builtin_amdgcn_wmma_f32_16x16x4_f32(false, a, false, b, (short)0, c,
                                               false, false);
}

// One wave (32 threads) per block; block handles 16 sequences.
__global__ __launch_bounds__(32) void farnn_main(
    const int* __restrict__ x, const int* __restrict__ lengths,
    const float* __restrict__ ER, const float* __restrict__ ws,
    const float* __restrict__ D2, const float* __restrict__ W1,
    const float* __restrict__ b1, const float* __restrict__ W2,
    const float* __restrict__ b2, const float* __restrict__ W3,
    const float* __restrict__ b3, float* __restrict__ out) {
  __shared__ int   tokbuf[16];
  __shared__ float Bbuf[16][RDIM];      // beta (C layout spill)  2 KB
  __shared__ float Ebuf[16][RDIM];      // gathered embeddings    2 KB
  __shared__ float alphaF[16][KDIM];    // final alpha           16 KB
  __shared__ float Hb1[16][H1D];
  __shared__ float Hb2[16][H2D];

  const int l    = threadIdx.x;
  const int lm   = l & 15;
  const int half = l >> 4;
  const int seq0 = blockIdx.x * 16;

  // Per-row activity lengths in both fragment layouts.
  int lenB = lengths[seq0 + lm];              // A-layout rows: m = lane%16
  int lenA[8];                                // C/D-layout rows: m = v + 8*half
#pragma unroll
  for (int v = 0; v < 8; ++v) lenA[v] = lengths[seq0 + v + 8 * half];

  // Persistent M^T B-fragments (32 VGPRs): B[k][n] = ws[k*32+n]
  v2f mfrag[2][8];
#pragma unroll
  for (int nt = 0; nt < 2; ++nt)
#pragma unroll
    for (int kc = 0; kc < 8; ++kc) {
      int n = 16 * nt + lm, k0 = 4 * kc + 2 * half;
      v2f f; f.x = ws[k0 * RDIM + n]; f.y = ws[(k0 + 1) * RDIM + n];
      mfrag[nt][kc] = f;
    }

  // beta in C/D layout: rows identical at init -> splat beta0[r]
  v8f betaC[2];
#pragma unroll
  for (int nt = 0; nt < 2; ++nt) {
    float b0 = ws[RDIM * RDIM + 16 * nt + lm];
#pragma unroll
    for (int v = 0; v < 8; ++v) betaC[nt][v] = b0;
  }
  v2f gam[8];                                 // gamma in A layout
#pragma unroll
  for (int kc = 0; kc < 8; ++kc) { gam[kc].x = 0.f; gam[kc].y = 0.f; }

  // ---- pipeline prologue: tokens(0) -> LDS, issue e(0), load tokens(1)
  int rtok = 0;
  if (l < 16) { rtok = x[(seq0 + l) * LSEQ + 0]; tokbuf[l] = rtok; }
  __syncthreads();
  v4f ereg[4];
#pragma unroll
  for (int i = 0; i < 4; ++i) {
    int idx = i * 32 + l, m = idx >> 3, q = idx & 7;
    ereg[i] = *(const v4f*)(ER + (size_t)tokbuf[m] * RDIM + q * 4);
  }
  if (l < 16) rtok = x[(seq0 + l) * LSEQ + 1];

#pragma unroll 1
  for (int t = 0; t < LSEQ; ++t) {
    // publish beta(t) and e(t) to LDS; publish tokens(t+1)
#pragma unroll
    for (int nt = 0; nt < 2; ++nt)
#pragma unroll
      for (int v = 0; v < 8; ++v)
        Bbuf[v + 8 * half][16 * nt + lm] = betaC[nt][v];
#pragma unroll
    for (int i = 0; i < 4; ++i) {
      int idx = i * 32 + l, m = idx >> 3, q = idx & 7;
      *(v4f*)(&Ebuf[m][q * 4]) = ereg[i];
    }
    if (l < 16) tokbuf[l] = rtok;
    __syncthreads();

    // prefetch e(t+1) into registers, tokens(t+2) into rtok
    v4f eregn[4];
#pragma unroll
    for (int i = 0; i < 4; ++i) { eregn[i].x = 0.f; eregn[i].y = 0.f; eregn[i].z = 0.f; eregn[i].w = 0.f; }
    int rtokn = 0;
    if (t + 1 < LSEQ) {
#pragma unroll
      for (int i = 0; i < 4; ++i) {
        int idx = i * 32 + l, m = idx >> 3, q = idx & 7;
        eregn[i] = *(const v4f*)(ER + (size_t)tokbuf[m] * RDIM + q * 4);
      }
      if (l < 16 && t + 2 < LSEQ) rtokn = x[(seq0 + l) * LSEQ + t + 2];
    }

    // gamma(t) = active ? beta(t) * e(t) : gamma(t-1)   (A layout)
    const bool actB = (t < lenB);
#pragma unroll
    for (int kc = 0; kc < 8; ++kc) {
      int r = 4 * kc + 2 * half;
      v2f bb = *(const v2f*)(&Bbuf[lm][r]);
      v2f ee = *(const v2f*)(&Ebuf[lm][r]);
      float cx = bb.x * ee.x, cy = bb.y * ee.y;
      gam[kc].x = actB ? cx : gam[kc].x;
      gam[kc].y = actB ? cy : gam[kc].y;
    }

    // beta(t+1) = gamma(t) @ M^T  via 16x WMMA f32, 4 interleaved accumulators
    v8f z = {};
    v8f a00 = z, a01 = z, a10 = z, a11 = z;
#pragma unroll
    for (int kc = 0; kc < 8; kc += 2) {
      a00 = wmma_f32_4(gam[kc],     mfrag[0][kc],     a00);
      a10 = wmma_f32_4(gam[kc],     mfrag[1][kc],     a10);
      a01 = wmma_f32_4(gam[kc + 1], mfrag[0][kc + 1], a01);
      a11 = wmma_f32_4(gam[kc + 1], mfrag[1][kc + 1], a11);
    }
    v8f nb0 = a00 + a01, nb1 = a10 + a11;
#pragma unroll
    for (int v = 0; v < 8; ++v) {
      bool a = (t < lenA[v]);
      betaC[0][v] = a ? nb0[v] : betaC[0][v];
      betaC[1][v] = a ? nb1[v] : betaC[1][v];
    }
#pragma unroll
    for (int i = 0; i < 4; ++i) ereg[i] = eregn[i];
    rtok = rtokn;
  }

  // alpha_final = gamma_last @ D2^T  (16x256), pure WMMA, B streamed from global
#pragma unroll 4
  for (int nt2 = 0; nt2 < 16; ++nt2) {
    v8f accF = {};
    int n = 16 * nt2 + lm;
#pragma unroll
    for (int kc = 0; kc < 8; ++kc) {
      int k0 = 4 * kc + 2 * half;
      v2f bf = *(const v2f*)(D2 + (size_t)n * RDIM + k0);  // D2^T[k][n] = D2[n][k]
      accF = wmma_f32_4(gam[kc], bf, accF);
    }
#pragma unroll
    for (int v = 0; v < 8; ++v) alphaF[v + 8 * half][n] = accF[v];
  }
  __syncthreads();

  // ---- tiny MLP tail (VALU): 256 -> 100 -> 100 -> 10
  for (int o = l; o < 16 * H1D; o += 32) {
    int s = o / H1D, j = o % H1D;
    float acc = b1[j];
    for (int c = 0; c < KDIM; ++c) acc = fmaf(alphaF[s][c], W1[j * KDIM + c], acc);
    Hb1[s][j] = fmaxf(acc, 0.f);
  }
  __syncthreads();
  for (int o = l; o < 16 * H2D; o += 32) {
    int s = o / H2D, j = o % H2D;
    float acc = b2[j];
    for (int c = 0; c < H1D; ++c) acc = fmaf(Hb1[s][c], W2[j * H1D + c], acc);
    Hb2[s][j] = fmaxf(acc, 0.f);
  }
  __syncthreads();
  for (int o = l; o < 16 * NLAB; o += 32) {
    int s = o / NLAB, j = o % NLAB;
    float acc = b3[j];
    for (int c = 0; c < H2D; ++c) acc = fmaf(Hb2[s][c], W3[j * H2D + c], acc);
    out[(size_t)(seq0 + s) * NLAB + j] = 1.0f * acc;  // AMP = 1.0
  }
}

extern "C" void kernel_launch(void* const* d_in, const int* in_sizes, int n_in,
                              void* d_out, int out_size, void* d_ws, size_t ws_size,
                              hipStream_t stream) {
  const int*   x      = (const int*)d_in[0];
  const int*   lens   = (const int*)d_in[1];
  const float* alpha0 = (const float*)d_in[2];
  const float* ER     = (const float*)d_in[3];
  const float* D1     = (const float*)d_in[4];
  const float* D2     = (const float*)d_in[5];
  const float* W1     = (const float*)d_in[6];
  const float* b1     = (const float*)d_in[7];
  const float* W2     = (const float*)d_in[8];
  const float* b2     = (const float*)d_in[9];
  const float* W3     = (const float*)d_in[10];
  const float* b3     = (const float*)d_in[11];
  float* ws  = (float*)d_ws;
  float* out = (float*)d_out;

  farnn_setup<<<1, 1024, 0, stream>>>(alpha0, D1, D2, ws);
  farnn_main<<<2048 / 16, 32, 0, stream>>>(x, lens, ER, ws, D2, W1, b1, W2, b2,
                                           W3, b3, out);
}